// FullMultiEmbedding_8538394984706
// MI455X (gfx1250) — compile-verified
//
#include <hip/hip_runtime.h>

// FullMultiEmbedding for MI455X (gfx1250):
//   out[b, idx, :] = weight[idx, :] * psw[i]  for each (b, i, idx), rest zeros.
// BATCH=256, BAG=50, NUM_EMB=10000, D=64 -> out = 655 MB, payload = 3.3 MB.
// Pure store-bandwidth problem (~28 us floor at 23.3 TB/s). Plan:
//   K1: non-temporal B128 zero-fill, 4x-unrolled grid-stride loop so store
//       issue is not throttled by 64-bit index bookkeeping.
//   K2: one WG per bag; async global->LDS B128 staging of the bag's rows
//       (ASYNCcnt path + s_wait_asynccnt), scale in-register, NT B128 scatter.

#define EMB_D      64     // floats per embedding row
#define ROWS_CHUNK 64     // max rows staged in LDS per pass (50 fits in one)

typedef float v4f __attribute__((ext_vector_type(4)));

// ---------------------------------------------------------------------------
// Kernel 1: zero-fill d_out with wide non-temporal stores.
// Output (655 MB) >> L2 (192 MB) -> NT stores avoid cache thrash.
// 4x unroll: one bounds check per 4 stores; stores stay fully coalesced
// (lane-contiguous B128, groups separated by the full grid stride).
// ---------------------------------------------------------------------------
__global__ __launch_bounds__(256) void femb_fill_zero(v4f* __restrict__ out,
                                                      long long n4) {
  const long long tid    = (long long)blockIdx.x * blockDim.x + threadIdx.x;
  const long long stride = (long long)gridDim.x * blockDim.x;
  const v4f z = {0.0f, 0.0f, 0.0f, 0.0f};

  long long i = tid;
  // main body: 4 NT B128 stores per trip
  for (; i + 3 * stride < n4; i += 4 * stride) {
    __builtin_nontemporal_store(z, out + i);
    __builtin_nontemporal_store(z, out + i + stride);
    __builtin_nontemporal_store(z, out + i + 2 * stride);
    __builtin_nontemporal_store(z, out + i + 3 * stride);
  }
  // tail
  for (; i < n4; i += stride) {
    __builtin_nontemporal_store(z, out + i);
  }
}

// ---------------------------------------------------------------------------
// Kernel 2: one workgroup (8 wave32s) per bag.
// Stage rows via CDNA5 async global->LDS B128 copies, then scale + scatter.
// Thread t handles float4-chunk (t&15) of rows (t>>4), (t>>4)+16, ... — the
// same mapping for stage and consume, so s_wait_asynccnt 0 covers the data
// each wave reads back; barriers kept for LDS reuse across chunks.
// ---------------------------------------------------------------------------
__global__ __launch_bounds__(256) void femb_scatter(
    const int*   __restrict__ input,    // [total] flat indices
    const int*   __restrict__ offsets,  // [B] bag starts
    const float* __restrict__ psw,      // [total] per-sample weights
    const float* __restrict__ weight,   // [num_emb, 64]
    float*       __restrict__ out,      // [B, num_emb, 64]
    int B, int total, int num_emb)
{
  __shared__ float smem[ROWS_CHUNK * EMB_D];   // 16 KB of the 320 KB/WGP LDS

  const int b     = blockIdx.x;
  const int t     = threadIdx.x;
  const int start = offsets[b];
  const int end   = (b + 1 < B) ? offsets[b + 1] : total;

  const int chunk = t & 15;   // which float4 of the 64-float row
  const int rbase = t >> 4;   // row slot within a 16-row pass

  // Low 32 bits of a generic LDS pointer == LDS byte offset (aperture in hi32).
  const unsigned lds_base = (unsigned)(uintptr_t)(&smem[0]);

  for (int base = start; base < end; base += ROWS_CHUNK) {
    const int cnt = min(ROWS_CHUNK, end - base);

    // Up to 4 passes of 16 rows each (ROWS_CHUNK=64). Preload index + weight
    // for this thread's rows so the scalar loads overlap the async copies.
    int   my_idx[4];
    float my_w[4];
    const int npass = (cnt + 15) >> 4;
#pragma unroll
    for (int p = 0; p < 4; ++p) {
      const int r = p * 16 + rbase;
      if (p < npass && r < cnt) {
        my_idx[p] = input[base + r];
        my_w[p]   = psw[base + r];
      } else {
        my_idx[p] = 0;
        my_w[p]   = 0.0f;
      }
    }

    // ---- stage: async global -> LDS (ASYNCcnt), 16 B per lane per row-chunk
#pragma unroll
    for (int p = 0; p < 4; ++p) {
      const int r = p * 16 + rbase;
      if (p < npass && r < cnt) {
        const unsigned lds_off =
            lds_base + (unsigned)((r * EMB_D + chunk * 4) * sizeof(float));
        const unsigned long long gaddr =
            (unsigned long long)(uintptr_t)(weight +
                                            (size_t)my_idx[p] * EMB_D +
                                            (size_t)chunk * 4);
        asm volatile("global_load_async_to_lds_b128 %0, %1, off"
                     :: "v"(lds_off), "v"(gaddr)
                     : "memory");
      }
    }
    asm volatile("s_wait_asynccnt 0x0" ::: "memory");
    __syncthreads();

    // ---- consume: scale by per-sample weight, NT scatter to dense output
#pragma unroll
    for (int p = 0; p < 4; ++p) {
      const int r = p * 16 + rbase;
      if (p < npass && r < cnt) {
        v4f v = *(const v4f*)&smem[r * EMB_D + chunk * 4];
        v *= my_w[p];
        v4f* dst = (v4f*)(out +
                          ((size_t)b * num_emb + (size_t)my_idx[p]) * EMB_D +
                          (size_t)chunk * 4);
        __builtin_nontemporal_store(v, dst);
      }
    }
    __syncthreads();  // protect LDS reuse across chunks (cnt > 64 case)
  }
}

// ---------------------------------------------------------------------------
// Launch: fill then scatter, stream-ordered.
// ---------------------------------------------------------------------------
extern "C" void kernel_launch(void* const* d_in, const int* in_sizes, int n_in,
                              void* d_out, int out_size, void* d_ws, size_t ws_size,
                              hipStream_t stream) {
  const int*   input   = (const int*)  d_in[0];  // input_ (indices)
  const int*   offsets = (const int*)  d_in[1];  // bag offsets
  const float* psw     = (const float*)d_in[2];  // per_sample_weights
  const float* weight  = (const float*)d_in[3];  // [num_emb, 64]
  float*       out     = (float*)d_out;

  const int total   = in_sizes[0];            // 12800
  const int B       = in_sizes[1];            // 256
  const int num_emb = in_sizes[3] / EMB_D;    // 10000

  // K1: zero-fill 655 MB with B128 NT stores.
  const long long n4 = (long long)out_size / 4;
  femb_fill_zero<<<dim3(8192), dim3(256), 0, stream>>>((v4f*)out, n4);

  // K2: one workgroup (8 wave32s) per bag.
  femb_scatter<<<dim3(B), dim3(256), 0, stream>>>(
      input, offsets, psw, weight, out, B, total, num_emb);
}